// FLIFP_59811714564167
// MI455X (gfx1250) — compile-verified
//
#include <hip/hip_runtime.h>
#include <math.h>

// ---------------------------------------------------------------------------
// FLIFP fused forward for MI455X (gfx1250, wave32, WMMA).
//
//   memory_V(N) = sum_{j=0}^{N-2} c[j] * hist[N-2-j],  c[j]=(j+3)^0.8-(j+2)^0.8
//
// Blocked by 16 time steps: far history -> V_WMMA_F32_16X16X4_F32 GEMM
// (Toeplitz(c) 16xK  x  hist Kx16-neurons):
//   * tap table stored REVERSED in LDS (cr[t] = c[1039-t]) so BOTH WMMA
//     operands are memory-ascending pairs -> pure ds_load_2addr, zero swap
//     movs, both walking pointers increment,
//   * two independent accumulator chains (even/odd chunks) to pipeline WMMA,
//   * register double-buffering so each WMMA waits on the PREVIOUS
//     iteration's LDS loads,
//   * block's 16 I-values preloaded into VGPRs so global latency overlaps
//     the GEMM instead of the serial tail.
// Near (<16 tap) recursive tail on VALU. History tile (1024x16 f32 = 64KB)
// resident in LDS for the whole scan.
// ---------------------------------------------------------------------------

typedef __attribute__((ext_vector_type(2))) float v2f;
typedef __attribute__((ext_vector_type(8))) float v8f;

#define T_STEPS 1024
#define L_DIM   1024
#define B_DIM   8
#define NTILE   16
#define C_PAD   1040           // reversed taps: cr[t] = c[1039-t]; cr[0..17]=0 pad
#define CR_TOP  (C_PAD - 1)    // 1039; cr[1039] = c[0]
#define LDS_FLOATS (T_STEPS * NTILE + C_PAD)

__launch_bounds__(32)
__global__ void flifp_fused_kernel(const float* __restrict__ I,
                                   float* __restrict__ out_spk,
                                   float* __restrict__ out_v,
                                   float coef)
{
    extern __shared__ float smem[];
    float* hist_lds = smem;                    // [k][n]  64 KB
    float* cr_lds   = smem + T_STEPS * NTILE;  // reversed taps, ~4 KB

    const int lane    = threadIdx.x & 31;
    const int n_local = lane & 15;                    // neuron within tile
    const int neuron  = blockIdx.x * NTILE + n_local; // global neuron (b*L + l)
    const int b_idx   = neuron >> 10;
    const int l_idx   = neuron & (L_DIM - 1);

    // ---- init LDS: zero history, build reversed taps (double pow ~ numpy) ----
    for (int i = lane; i < T_STEPS * NTILE; i += 32) hist_lds[i] = 0.0f;
    for (int t = lane; t < C_PAD; t += 32) {
        // cr[t] = c[1039-t] = (1042-t)^0.8 - (1041-t)^0.8 for valid j<=1021
        float cj = 0.0f;
        if (t >= C_PAD - (T_STEPS - 2))  // t >= 18
            cj = (float)(pow((double)(1042 - t), 0.8) - pow((double)(1041 - t), 0.8));
        cr_lds[t] = cj;
    }
    __syncthreads();  // single wave: ordering only

    const float V0     = -70.0f;   // V_INIT / VL / V_RESET
    const float GLc    = 0.025f;
    const float VLc    = -70.0f;
    const float THR    = -50.0f;
    const float RESETV = 20.0f;    // THRESHOLD - V_RESET

    // near-tail taps c[0..14] = cr[1039-j] in registers (wave-uniform)
    float cnear[15];
#pragma unroll
    for (int j = 0; j < 15; ++j) cnear[j] = cr_lds[CR_TOP - j];

    const size_t plane = ((size_t)b_idx * T_STEPS) * L_DIM + l_idx; // + N*L

    // ---- steps 0 and 1 (special-cased in reference) ----
    if (lane < 16) {
        out_v[plane]   = V0;
        out_spk[plane] = 0.0f;                  // heaviside(-70 + 50) = 0
    }
    const float I0 = I[plane];                  // I[b, 0, l]
    const float V1 = V0 + 0.005f * (-V0 + I0 * 40.0f); // DT/TAU=0.005, 1/GL=40
    if (lane < 16) {
        hist_lds[n_local] = V1 - V0;            // hist[0]
        const size_t off1 = plane + L_DIM;
        out_v[off1]   = V1;
        out_spk[off1] = (V1 - THR) > 0.0f ? 1.0f : 0.0f;
    }
    float V_old = V1;

    const int kkA = (lane >> 4) << 1;           // K sub-index 0 or 2 (WMMA layout)

    // ---- main blocked scan: N in [N0, N0+15] ----
    for (int N0 = 2; N0 < T_STEPS; N0 += 16) {
        if (N0 + 16 < T_STEPS)                  // stream next I block toward L2/L0
            __builtin_prefetch(&I[plane + (size_t)(N0 + 16) * L_DIM], 0, 1);

        // Preload this block's 16 I values; loads overlap the whole GEMM.
        float Ireg[16];
#pragma unroll
        for (int m = 0; m < 16; ++m) {
            const int Nc = (N0 + m < T_STEPS) ? (N0 + m) : (T_STEPS - 1);
            Ireg[m] = I[plane + (size_t)Nc * L_DIM];
        }

        // Far-history GEMM over k in [0, N0-2]:
        //   acc[m][n] = sum_k c[N0+m-2-k] * hist[k][n]
        // With reversed taps, chunk ch (kbase=4ch), K-slot kk:
        //   A[m][kk] = cr[(1039-(N0-2)) - m + kbase + kk]   (ascending pair)
        //   B[kk][n] = hist[kbase+kk][n]                    (ascending rows)
        const int fullChunks = (N0 - 1) >> 2;
        const int rem        = (N0 - 1) & 3;
        const int pairs      = fullChunks >> 1;

        v8f acc0 = {0.f, 0.f, 0.f, 0.f, 0.f, 0.f, 0.f, 0.f};
        v8f acc1 = {0.f, 0.f, 0.f, 0.f, 0.f, 0.f, 0.f, 0.f};

        const float* pa = cr_lds + (CR_TOP - (N0 - 2) - n_local + kkA); // chunk 0
        const float* pb = hist_lds + kkA * NTILE + n_local;             // chunk 0

        v2f a0c, b0c, a1c, b1c;
        if (pairs > 0) {                        // preload pair 0
            a0c.x = pa[0];         a0c.y = pa[1];
            b0c.x = pb[0];         b0c.y = pb[NTILE];
            a1c.x = pa[4];         a1c.y = pa[5];
            b1c.x = pb[4 * NTILE]; b1c.y = pb[5 * NTILE];
        }
#pragma unroll 2
        for (int it = 1; it < pairs; ++it) {    // load pair it, MMA pair it-1
            pa += 8;
            pb += 8 * NTILE;
            v2f a0n, b0n, a1n, b1n;
            a0n.x = pa[0];         a0n.y = pa[1];
            b0n.x = pb[0];         b0n.y = pb[NTILE];
            a1n.x = pa[4];         a1n.y = pa[5];
            b1n.x = pb[4 * NTILE]; b1n.y = pb[5 * NTILE];
            acc0 = __builtin_amdgcn_wmma_f32_16x16x4_f32(
                     false, a0c, false, b0c, (short)0, acc0, false, false);
            acc1 = __builtin_amdgcn_wmma_f32_16x16x4_f32(
                     false, a1c, false, b1c, (short)0, acc1, false, false);
            a0c = a0n; b0c = b0n; a1c = a1n; b1c = b1n;
        }
        if (pairs > 0) {                        // drain last pair
            acc0 = __builtin_amdgcn_wmma_f32_16x16x4_f32(
                     false, a0c, false, b0c, (short)0, acc0, false, false);
            acc1 = __builtin_amdgcn_wmma_f32_16x16x4_f32(
                     false, a1c, false, b1c, (short)0, acc1, false, false);
            pa += 8;
            pb += 8 * NTILE;
        }
        if (fullChunks & 1) {                   // leftover full chunk
            v2f a, b;
            a.x = pa[0];  a.y = pa[1];
            b.x = pb[0];  b.y = pb[NTILE];
            acc1 = __builtin_amdgcn_wmma_f32_16x16x4_f32(
                     false, a, false, b, (short)0, acc1, false, false);
        }
        if (rem) {                              // peeled remainder: clamp cr idx
            const int kbase = fullChunks << 2;  // up; overshoot B rows read zeros
            int ir0 = CR_TOP - (rem - 1) - n_local + kkA + kbase - kbase; // see note
            ir0 = CR_TOP - ((rem - 1) + n_local - kkA);
            if (ir0 > CR_TOP) ir0 = CR_TOP;
            int ir1 = ir0 + 1;
            if (ir1 > CR_TOP) ir1 = CR_TOP;
            v2f a, b;
            a.x = cr_lds[ir0];
            a.y = cr_lds[ir1];
            b.x = hist_lds[(kbase + kkA) * NTILE + n_local];
            b.y = hist_lds[(kbase + kkA + 1) * NTILE + n_local];
            acc0 = __builtin_amdgcn_wmma_f32_16x16x4_f32(
                     false, a, false, b, (short)0, acc0, false, false);
        }
        const v8f acc = acc0 + acc1;

        // Sequential 16-step tail (near history in registers)
        float dloc[16];
#pragma unroll
        for (int m = 0; m < 16; ++m) {
            const int N = N0 + m;
            if (N < T_STEPS) {                  // uniform guard (last block: 14 steps)
                // far[m][n]: m<8 -> lane n elem m ; m>=8 -> lane n+16 elem m-8
                const float farm =
                    __shfl(acc[m & 7], n_local + ((m < 8) ? 0 : 16), 32);
                float nearv = 0.0f;
#pragma unroll
                for (int j = 0; j < m; ++j)
                    nearv = fmaf(cnear[j], dloc[m - 1 - j], nearv);
                const float gate = (V_old - THR) > 0.0f ? 1.0f : 0.0f;
                const float Vpre =
                    fmaf(coef, fmaf(-GLc, (V_old - VLc), Ireg[m]), V_old) - farm - nearv;
                const float dm = Vpre - V_old;
                dloc[m] = dm;
                const float Vpost = Vpre - gate * RESETV;
                if (lane < 16) {
                    hist_lds[(N - 1) * NTILE + n_local] = dm;   // hist[N-1]
                    const size_t off = plane + (size_t)N * L_DIM;
                    out_v[off]   = Vpost;
                    out_spk[off] = (Vpost - THR) > 0.0f ? 1.0f : 0.0f;
                }
                V_old = Vpost;
            }
        }
        // LDS writes above are in-order with next block's LDS reads (same wave).
    }
}

extern "C" void kernel_launch(void* const* d_in, const int* in_sizes, int n_in,
                              void* d_out, int out_size, void* d_ws, size_t ws_size,
                              hipStream_t stream) {
    (void)in_sizes; (void)n_in; (void)out_size; (void)d_ws; (void)ws_size;
    const float* I = (const float*)d_in[0];           // [8, 1024, 1024] f32
    float* out = (float*)d_out;
    const size_t plane_elems = (size_t)B_DIM * T_STEPS * L_DIM;
    float* out_spk = out;                             // spikes  [B,T,L]
    float* out_v   = out + plane_elems;               // V trace [B,T,L]

    // coef = DT^alpha * Gamma(2-alpha) / CM  (host double, deterministic)
    const float coef = (float)(pow(0.1, 0.2) * tgamma(1.8) / 0.5);

    dim3 grid((B_DIM * L_DIM) / NTILE);               // 512 tiles, 1 wave each
    dim3 block(32);
    flifp_fused_kernel<<<grid, block, LDS_FLOATS * sizeof(float), stream>>>(
        I, out_spk, out_v, coef);
}